// DigitCap_30133490548808
// MI455X (gfx1250) — compile-verified
//
#include <hip/hip_runtime.h>
#include <math.h>

typedef __attribute__((ext_vector_type(2))) float v2f;
typedef __attribute__((ext_vector_type(8))) float v8f;

#define NPRIM 512
#define DP    8
#define DD    16
#define NDIG  10
#define BATCH 64
#define KDIM  (NPRIM * DP)   // 4096
#define NOUT  (NDIG * DD)    // 160
#define EPS   1e-7f

// ---------------------------------------------------------------------------
// Kernel 1: fold the routing prior (1 + B[d,n]) into the transform tensor and
// transpose W[d,n,i,j] -> Wt[(d*16+i)][(n*8+j)]  (row-major 160 x 4096),
// which is exactly the B^T operand of the collapsed GEMM.
// (Softmax over the size-1 axis in the reference is identically 1, so the
//  whole Gram/attention stage cancels exactly; only this GEMM remains.)
// ---------------------------------------------------------------------------
__global__ __launch_bounds__(256)
void prep_weights(const float* __restrict__ W,
                  const float* __restrict__ Bp,
                  float* __restrict__ Wt) {
  int idx = blockIdx.x * 256 + threadIdx.x;         // over 10*512*16*8 = 655360
  if (idx >= NDIG * NPRIM * DD * DP) return;
  int j = idx & 7;
  int i = (idx >> 3) & 15;
  int n = (idx >> 7) & 511;
  int d = idx >> 16;
  float scale = 1.0f + Bp[d * NPRIM + n];
  Wt[(d * DD + i) * KDIM + n * DP + j] = W[idx] * scale;
}

// ---------------------------------------------------------------------------
// Kernel 2: S = U (64x4096) @ Wt^T (4096x160) via V_WMMA_F32_16X16X4_F32,
// then capsule squash per (batch, digit) 16-vector.
//
// Fragment layout (ISA 7.12.2, f32 16x4 A / 4x16 B): lane%16 = M (A) / N (B),
// lane/16 selects K in {0,1} vs {2,3}, VGPR index = K lsb.  The K-half select
// is folded into the per-lane base pointer (+2*half), so each step loads its
// exact v2f fragment with one global_load_b64 — no selection VALU in the loop,
// and every loaded byte is consumed exactly once.
//
// Block = 128 threads (4 waves); each block owns one 16x16 output tile,
// each wave a 1024-wide K slice; partials reduced through LDS.
// ---------------------------------------------------------------------------
__global__ __launch_bounds__(128)
void caps_gemm_squash(const float* __restrict__ U,    // [64][4096]
                      const float* __restrict__ Wt,   // [160][4096]
                      float* __restrict__ out) {      // [64][10][16]
  __shared__ float part[4][8][32];

  const int lane   = threadIdx.x & 31;
  const int wave   = threadIdx.x >> 5;
  const int tile_m = blockIdx.x & 3;     // 4 tiles over batch (64/16)
  const int d      = blockIdx.x >> 2;    // 10 digit-caps tiles (grid.x = 40)
  const int half   = lane >> 4;          // K-half selector per ISA A/B layout
  const int lm     = lane & 15;          // M for A-frag, N for B-frag

  const float* __restrict__ ap =
      U  + (size_t)(tile_m * 16 + lm) * KDIM + 2 * half;
  const float* __restrict__ bp =
      Wt + (size_t)(d * DD + lm) * KDIM + 2 * half;

  v8f acc0 = {0.f, 0.f, 0.f, 0.f, 0.f, 0.f, 0.f, 0.f};
  v8f acc1 = {0.f, 0.f, 0.f, 0.f, 0.f, 0.f, 0.f, 0.f};

#define WSTEP(af, bf, accv)                                                    \
  accv = __builtin_amdgcn_wmma_f32_16x16x4_f32(false, (af), false, (bf),       \
                                               (short)0, (accv), false, false)

  const int k0 = wave * (KDIM / 4);
  for (int kb = k0; kb < k0 + KDIM / 4; kb += 16) {
    v2f a0 = *(const v2f*)(ap + kb);
    v2f a1 = *(const v2f*)(ap + kb + 4);
    v2f a2 = *(const v2f*)(ap + kb + 8);
    v2f a3 = *(const v2f*)(ap + kb + 12);
    v2f b0 = *(const v2f*)(bp + kb);
    v2f b1 = *(const v2f*)(bp + kb + 4);
    v2f b2 = *(const v2f*)(bp + kb + 8);
    v2f b3 = *(const v2f*)(bp + kb + 12);
    WSTEP(a0, b0, acc0);
    WSTEP(a1, b1, acc1);
    WSTEP(a2, b2, acc0);
    WSTEP(a3, b3, acc1);
  }
#undef WSTEP

  // Stage this wave's partial 16x16 tile in LDS.
#pragma unroll
  for (int r = 0; r < 8; ++r) part[wave][r][lane] = acc0[r] + acc1[r];
  __syncthreads();

  if (wave == 0) {
    // C/D layout: VGPR r, lanes 0-15 -> M=r, N=lane; lanes 16-31 -> M=r+8.
#pragma unroll
    for (int r = 0; r < 8; ++r) {
      float s = part[0][r][lane] + part[1][r][lane] +
                part[2][r][lane] + part[3][r][lane];
      // ||S[b,d,:]|| : reduce squares across the 16 lanes of this half.
      float sq = s * s;
      sq += __shfl_xor(sq, 1, 32);
      sq += __shfl_xor(sq, 2, 32);
      sq += __shfl_xor(sq, 4, 32);
      sq += __shfl_xor(sq, 8, 32);
      float norm = sqrtf(sq);
      float g = (1.0f - __expf(-norm)) / (norm + EPS);
      int m = tile_m * 16 + r + 8 * half;          // batch index
      int i = lane & 15;                           // digit-cap component
      out[(m * NDIG + d) * DD + i] = s * g;
    }
  }
}

// ---------------------------------------------------------------------------
// Harness entry point.
// d_in[0] = primary_caps f32 [64,512,8]
// d_in[1] = W            f32 [10,512,16,8]
// d_in[2] = B            f32 [10,1,512]
// d_out   = f32 [64,10,16]
// d_ws    : 160*4096*4 = 2.62 MB used for the folded/transposed weights.
// ---------------------------------------------------------------------------
extern "C" void kernel_launch(void* const* d_in, const int* in_sizes, int n_in,
                              void* d_out, int out_size, void* d_ws,
                              size_t ws_size, hipStream_t stream) {
  const float* U  = (const float*)d_in[0];
  const float* W  = (const float*)d_in[1];
  const float* Bp = (const float*)d_in[2];
  float* Wt = (float*)d_ws;

  const int wtotal = NDIG * NPRIM * DD * DP;       // 655360
  prep_weights<<<(wtotal + 255) / 256, 256, 0, stream>>>(W, Bp, Wt);
  caps_gemm_squash<<<4 * NDIG, 128, 0, stream>>>(U, Wt, (float*)d_out);
}